// FECI_Mamba_910533066821
// MI455X (gfx1250) — compile-verified
//
#include <hip/hip_runtime.h>

// ---------------------------------------------------------------------------
// Cross-Mamba stack for MI455X (gfx1250, wave32).
//  - Projections on the fp32 matrix pipe: V_WMMA_F32_16X16X4_F32.
//  - Selective scan: 3-pass chunked scan (chunk aggregate of the decay is the
//    closed form exp(A * sum(dt))), 16x the parallelism of a sequential scan,
//    exp() on the TRANS pipe (co-executes with VALU on CDNA5).
//  - All intermediates (~87MB) stay in d_ws => resident in the 192MB L2.
// ---------------------------------------------------------------------------

#define D_MODEL 64
#define D_INNER 128
#define D_STATE 16
#define D_CONV  4
#define DT_RANK 4
#define NLAYERS 4
#define BB      8
#define LL      2048
#define M_TOK   (BB * LL)                  // 16384 tokens
#define XD_ROWS (DT_RANK + 2 * D_STATE)    // 36
#define NCHUNK  16
#define CLEN    (LL / NCHUNK)              // 128

typedef float v2f __attribute__((ext_vector_type(2)));
typedef float v8f __attribute__((ext_vector_type(8)));

__device__ __forceinline__ float sigmoidf_(float x) { return 1.0f / (1.0f + __expf(-x)); }
__device__ __forceinline__ float siluf_(float x)    { return x * sigmoidf_(x); }
__device__ __forceinline__ float softplusf_(float x){ return (x > 20.0f) ? x : log1pf(__expf(x)); }

// ---------------------------------------------------------------------------
// LayerNorm over D_MODEL=64: one wave per token, 2 elements per lane.
// ---------------------------------------------------------------------------
template <bool ADD2>
__global__ void ln_kernel(const float* __restrict__ xa, const float* __restrict__ xb,
                          const float* __restrict__ w, const float* __restrict__ bia,
                          float* __restrict__ out)
{
    int token = blockIdx.x * (blockDim.x >> 5) + (threadIdx.x >> 5);
    int lane  = threadIdx.x & 31;
    size_t base = (size_t)token * D_MODEL;

    float v0 = xa[base + lane];
    float v1 = xa[base + lane + 32];
    if (ADD2) { v0 += xb[base + lane]; v1 += xb[base + lane + 32]; }

    float s  = v0 + v1;
    float ss = v0 * v0 + v1 * v1;
#pragma unroll
    for (int off = 16; off >= 1; off >>= 1) {
        s  += __shfl_xor(s, off, 32);
        ss += __shfl_xor(ss, off, 32);
    }
    float mean = s * (1.0f / 64.0f);
    float var  = ss * (1.0f / 64.0f) - mean * mean;
    float rstd = rsqrtf(var + 1e-5f);

    out[base + lane]      = (v0 - mean) * rstd * w[lane]      + bia[lane];
    out[base + lane + 32] = (v1 - mean) * rstd * w[lane + 32] + bia[lane + 32];
}

// ---------------------------------------------------------------------------
// WMMA GEMM: Out[m, n] = sum_k X[m, k] * W[n, k]
//   X token-major:   X[m*CBx + k]                 (CBx == K)
//   X channel-major: X[(b*CBx + k)*LL + l], m = b*LL + l
//   Store channel-major: Out[(b*CBout + n)*LL + l]
//   Store token+relu:    Out[m*N + n] = relu(acc + Resid[m*N + n])
// 4 waves per block; each wave computes one 16x16 tile (its own N-tile)
// via V_WMMA_F32_16X16X4_F32.
// ---------------------------------------------------------------------------
template <int K, bool X_TOKEN_MAJOR, bool ADD_X2, bool STORE_TOKEN_RELU>
__global__ void gemm_wmma(const float* __restrict__ X, const float* __restrict__ X2,
                          const float* __restrict__ W, float* __restrict__ Out,
                          const float* __restrict__ Resid,
                          int N, int CBx, int CBout)
{
    const int wave = threadIdx.x >> 5;
    const int lid  = threadIdx.x & 31;
    const int m0   = blockIdx.x * 16;
    const int n0   = (blockIdx.y * 4 + wave) * 16;
    const int half = lid >> 4;       // 0: K=k0..k0+1 ; 1: K=k0+2..k0+3
    const int lm   = lid & 15;
    const int m    = m0 + lm;        // A row (token)
    const int n    = n0 + lm;        // B col
    const int nw   = (n < N) ? n : 0;

    const int b = m >> 11;           // LL = 2048
    const int l = m & (LL - 1);

    v8f acc = {};
#pragma unroll
    for (int k0 = 0; k0 < K; k0 += 4) {
        const int ka = k0 + 2 * half;
        v2f afrag, bfrag;
        if (X_TOKEN_MAJOR) {
            const float* xp = X + (size_t)m * CBx + ka;
            afrag.x = xp[0]; afrag.y = xp[1];
            if (ADD_X2) {
                const float* x2p = X2 + (size_t)m * CBx + ka;
                afrag.x += x2p[0]; afrag.y += x2p[1];
            }
        } else {
            const float* xp = X + ((size_t)(b * CBx + ka)) * LL + l;
            afrag.x = xp[0]; afrag.y = xp[LL];
            if (ADD_X2) {
                const float* x2p = X2 + ((size_t)(b * CBx + ka)) * LL + l;
                afrag.x += x2p[0]; afrag.y += x2p[LL];
            }
        }
        const float* wp = W + (size_t)nw * K + ka;
        bfrag.x = wp[0]; bfrag.y = wp[1];

        acc = __builtin_amdgcn_wmma_f32_16x16x4_f32(
            false, afrag, false, bfrag, (short)0, acc, false, false);
    }

#pragma unroll
    for (int v = 0; v < 8; ++v) {
        const int mr = m0 + half * 8 + v;   // D row (token)
        const int nc = n0 + lm;             // D col
        if (nc < N) {
            if (STORE_TOKEN_RELU) {
                float val = acc[v] + Resid[(size_t)mr * N + nc];
                Out[(size_t)mr * N + nc] = fmaxf(val, 0.0f);
            } else {
                const int br = mr >> 11;
                const int lr = mr & (LL - 1);
                Out[((size_t)(br * CBout + nc)) * LL + lr] = acc[v];
            }
        }
    }
}

// ---------------------------------------------------------------------------
// Depthwise causal conv (K=4) + SiLU, channel-major in/out.
// ---------------------------------------------------------------------------
__global__ void conv_silu_kernel(const float* __restrict__ in, const float* __restrict__ w,
                                 const float* __restrict__ bias, float* __restrict__ out,
                                 int CBin)
{
    int idx = blockIdx.x * blockDim.x + threadIdx.x;
    if (idx >= BB * D_INNER * LL) return;
    const int l = idx & (LL - 1);
    const int d = (idx >> 11) & (D_INNER - 1);
    const int b = idx >> 18;

    const float* ip = in + ((size_t)(b * CBin + d)) * LL;
    float acc = bias[d];
#pragma unroll
    for (int k = 0; k < D_CONV; ++k) {
        int ls = l - (D_CONV - 1) + k;
        if (ls >= 0) acc += w[d * D_CONV + k] * ip[ls];
    }
    out[((size_t)(b * D_INNER + d)) * LL + l] = siluf_(acc);
}

// ---------------------------------------------------------------------------
// Chunked selective scan, pass 1: per (b, d, chunk) run the 16-state
// recurrence from h=0 over CLEN steps; emit h_final[16] and sum(dt).
// ---------------------------------------------------------------------------
__global__ void scan_pass1(const float* __restrict__ u, const float* __restrict__ delta,
                           const float* __restrict__ xd, const float* __restrict__ A_log,
                           float* __restrict__ hagg, float* __restrict__ sdt)
{
    int tid = blockIdx.x * blockDim.x + threadIdx.x;
    if (tid >= BB * D_INNER * NCHUNK) return;
    const int c = tid & (NCHUNK - 1);
    const int d = (tid >> 4) & (D_INNER - 1);
    const int b = tid >> 11;

    float A[D_STATE], h[D_STATE];
#pragma unroll
    for (int n = 0; n < D_STATE; ++n) {
        A[n] = -__expf(A_log[d * D_STATE + n]);
        h[n] = 0.0f;
    }
    const size_t row = (size_t)(b * D_INNER + d) * LL + (size_t)c * CLEN;
    const float* dl = delta + row;
    const float* ul = u + row;
    const float* Bm = xd + ((size_t)(b * XD_ROWS + DT_RANK)) * LL + (size_t)c * CLEN;

    float S = 0.0f;
    for (int t = 0; t < CLEN; ++t) {
        const float dt = softplusf_(dl[t]);
        S += dt;
        const float du = dt * ul[t];
#pragma unroll
        for (int n = 0; n < D_STATE; ++n)
            h[n] = __expf(dt * A[n]) * h[n] + du * Bm[(size_t)n * LL + t];
    }
    float* hp = hagg + (size_t)tid * D_STATE;
#pragma unroll
    for (int n = 0; n < D_STATE; ++n) hp[n] = h[n];
    sdt[tid] = S;
}

// ---------------------------------------------------------------------------
// Pass 2: per (b, d) combine chunk aggregates; chunk decay product is the
// closed form exp(A[n]*sum(dt)). Overwrites hagg[c] with chunk c's h_init.
// ---------------------------------------------------------------------------
__global__ void scan_pass2(float* __restrict__ hagg, const float* __restrict__ sdt,
                           const float* __restrict__ A_log)
{
    int tid = blockIdx.x * blockDim.x + threadIdx.x;
    if (tid >= BB * D_INNER) return;
    const int d = tid & (D_INNER - 1);

    float A[D_STATE], h[D_STATE];
#pragma unroll
    for (int n = 0; n < D_STATE; ++n) {
        A[n] = -__expf(A_log[d * D_STATE + n]);
        h[n] = 0.0f;
    }
    for (int c = 0; c < NCHUNK; ++c) {
        float* hp = hagg + ((size_t)tid * NCHUNK + c) * D_STATE;
        const float g = __expf(A[0] * 0.0f); (void)g; // keep structure simple
        const float S = sdt[(size_t)tid * NCHUNK + c];
#pragma unroll
        for (int n = 0; n < D_STATE; ++n) {
            const float hf = hp[n];          // chunk-local final (from h=0)
            hp[n] = h[n];                    // overwrite with h_init for chunk c
            h[n] = __expf(A[n] * S) * h[n] + hf;
        }
    }
}

// ---------------------------------------------------------------------------
// Pass 3: per (b, d, chunk) re-run the recurrence from h_init, producing
// y_t = (sum_n h_n C_n + D*u) * silu(z).
// ---------------------------------------------------------------------------
__global__ void scan_pass3(const float* __restrict__ u, const float* __restrict__ delta,
                           const float* __restrict__ xd, const float* __restrict__ z,
                           const float* __restrict__ A_log, const float* __restrict__ Dp,
                           const float* __restrict__ hagg, float* __restrict__ y, int zCB)
{
    int tid = blockIdx.x * blockDim.x + threadIdx.x;
    if (tid >= BB * D_INNER * NCHUNK) return;
    const int c = tid & (NCHUNK - 1);
    const int d = (tid >> 4) & (D_INNER - 1);
    const int b = tid >> 11;

    float A[D_STATE], h[D_STATE];
    const float* hp = hagg + (size_t)tid * D_STATE;
#pragma unroll
    for (int n = 0; n < D_STATE; ++n) {
        A[n] = -__expf(A_log[d * D_STATE + n]);
        h[n] = hp[n];
    }
    const float Dd = Dp[d];
    const size_t row = (size_t)(b * D_INNER + d) * LL + (size_t)c * CLEN;
    const float* dl = delta + row;
    const float* ul = u + row;
    const float* zl = z + ((size_t)(b * zCB + d)) * LL + (size_t)c * CLEN;
    const float* Bm = xd + ((size_t)(b * XD_ROWS + DT_RANK)) * LL + (size_t)c * CLEN;
    const float* Cm = xd + ((size_t)(b * XD_ROWS + DT_RANK + D_STATE)) * LL + (size_t)c * CLEN;
    float* yl = y + row;

    for (int t = 0; t < CLEN; ++t) {
        const float dt = softplusf_(dl[t]);
        const float uv = ul[t];
        const float du = dt * uv;
        float accy = 0.0f;
#pragma unroll
        for (int n = 0; n < D_STATE; ++n) {
            h[n] = __expf(dt * A[n]) * h[n] + du * Bm[(size_t)n * LL + t];
            accy += h[n] * Cm[(size_t)n * LL + t];
        }
        yl[t] = (accy + Dd * uv) * siluf_(zl[t]);
    }
}

// ---------------------------------------------------------------------------
// Host side
// ---------------------------------------------------------------------------
struct Params {
    const float *in_proj_w, *in_proj1_w, *conv_w, *conv_b, *conv1_w, *conv1_b;
    const float *x_proj_w, *x_proj1_w, *dt_proj_w, *dt_proj1_w, *out_proj_w;
    const float *A_log, *A_log1, *Dv, *D1v;
};
struct Bufs {
    float *xz, *x1p, *co, *co1, *xd, *xd1, *dl0, *dl1, *y0, *y1;
    float *hag0, *hag1, *sdt0, *sdt1;
};

static inline dim3 gemm_grid(int N) { return dim3(M_TOK / 16, ((N + 15) / 16 + 3) / 4); }

static void run_scan(const float* co, const float* dl, const float* xd, const float* zbase,
                     const float* Al, const float* Dl, float* hag, float* sdt, float* y,
                     hipStream_t s)
{
    scan_pass1<<<(BB * D_INNER * NCHUNK) / 128, 128, 0, s>>>(co, dl, xd, Al, hag, sdt);
    scan_pass2<<<(BB * D_INNER) / 128, 128, 0, s>>>(hag, sdt, Al);
    scan_pass3<<<(BB * D_INNER * NCHUNK) / 128, 128, 0, s>>>(co, dl, xd, zbase, Al, Dl, hag, y,
                                                             2 * D_INNER);
}

static void cross_ssm(float* xio, const float* x1, int layer,
                      const Params& P, const Bufs& W, hipStream_t s)
{
    const float* ipw  = P.in_proj_w  + (size_t)layer * 2 * D_INNER * D_MODEL;
    const float* ipw1 = P.in_proj1_w + (size_t)layer * D_INNER * D_MODEL;
    const float* cw   = P.conv_w  + (size_t)layer * D_INNER * D_CONV;
    const float* cb   = P.conv_b  + (size_t)layer * D_INNER;
    const float* cw1  = P.conv1_w + (size_t)layer * D_INNER * D_CONV;
    const float* cb1  = P.conv1_b + (size_t)layer * D_INNER;
    const float* xpw  = P.x_proj_w  + (size_t)layer * XD_ROWS * D_INNER;
    const float* xpw1 = P.x_proj1_w + (size_t)layer * XD_ROWS * D_INNER;
    const float* dtw  = P.dt_proj_w  + (size_t)layer * D_INNER * DT_RANK;
    const float* dtw1 = P.dt_proj1_w + (size_t)layer * D_INNER * DT_RANK;
    const float* opw  = P.out_proj_w + (size_t)layer * D_MODEL * D_INNER;
    const float* Al   = P.A_log  + (size_t)layer * D_INNER * D_STATE;
    const float* Al1  = P.A_log1 + (size_t)layer * D_INNER * D_STATE;
    const float* Dl   = P.Dv  + (size_t)layer * D_INNER;
    const float* Dl1  = P.D1v + (size_t)layer * D_INNER;

    const dim3 blk(128);
    // in_proj: (B*L,64) x (256,64)^T -> xz channel-major (B,256,L)
    gemm_wmma<D_MODEL, true, false, false><<<gemm_grid(2 * D_INNER), blk, 0, s>>>(
        xio, nullptr, ipw, W.xz, nullptr, 2 * D_INNER, D_MODEL, 2 * D_INNER);
    // in_proj1 -> x1p (B,128,L)
    gemm_wmma<D_MODEL, true, false, false><<<gemm_grid(D_INNER), blk, 0, s>>>(
        x1, nullptr, ipw1, W.x1p, nullptr, D_INNER, D_MODEL, D_INNER);

    // causal conv + silu
    conv_silu_kernel<<<(BB * D_INNER * LL) / 256, 256, 0, s>>>(W.xz, cw, cb, W.co, 2 * D_INNER);
    conv_silu_kernel<<<(BB * D_INNER * LL) / 256, 256, 0, s>>>(W.x1p, cw1, cb1, W.co1, D_INNER);

    // x_proj: co (B,128,L) -> xd (B,36,L)
    gemm_wmma<D_INNER, false, false, false><<<gemm_grid(XD_ROWS), blk, 0, s>>>(
        W.co, nullptr, xpw, W.xd, nullptr, XD_ROWS, D_INNER, XD_ROWS);
    gemm_wmma<D_INNER, false, false, false><<<gemm_grid(XD_ROWS), blk, 0, s>>>(
        W.co1, nullptr, xpw1, W.xd1, nullptr, XD_ROWS, D_INNER, XD_ROWS);

    // dt_proj: xd rows 0..3 -> delta (B,128,L)
    gemm_wmma<DT_RANK, false, false, false><<<gemm_grid(D_INNER), blk, 0, s>>>(
        W.xd, nullptr, dtw, W.dl0, nullptr, D_INNER, XD_ROWS, D_INNER);
    gemm_wmma<DT_RANK, false, false, false><<<gemm_grid(D_INNER), blk, 0, s>>>(
        W.xd1, nullptr, dtw1, W.dl1, nullptr, D_INNER, XD_ROWS, D_INNER);

    // chunked selective scans (z lives at rows 128..255 of xz per batch)
    float* zbase = W.xz + (size_t)D_INNER * LL;
    run_scan(W.co,  W.dl0, W.xd,  zbase, Al,  Dl,  W.hag0, W.sdt0, W.y0, s);
    run_scan(W.co1, W.dl1, W.xd1, zbase, Al1, Dl1, W.hag1, W.sdt1, W.y1, s);

    // out_proj on (y0 + y1), add residual xio, relu, write xio in place
    gemm_wmma<D_INNER, false, true, true><<<gemm_grid(D_MODEL), blk, 0, s>>>(
        W.y0, W.y1, opw, xio, xio, D_MODEL, D_INNER, 0);
}

extern "C" void kernel_launch(void* const* d_in, const int* in_sizes, int n_in,
                              void* d_out, int out_size, void* d_ws, size_t ws_size,
                              hipStream_t stream)
{
    (void)in_sizes; (void)n_in; (void)out_size; (void)ws_size;

    const float* x_a   = (const float*)d_in[0];
    const float* x_b   = (const float*)d_in[1];
    const float* ln1_w = (const float*)d_in[2];
    const float* ln1_b = (const float*)d_in[3];
    const float* ln2_w = (const float*)d_in[4];
    const float* ln2_b = (const float*)d_in[5];
    const float* ln3_w = (const float*)d_in[6];
    const float* ln3_b = (const float*)d_in[7];

    Params P;
    P.in_proj_w  = (const float*)d_in[8];
    P.in_proj1_w = (const float*)d_in[9];
    P.conv_w     = (const float*)d_in[10];
    P.conv_b     = (const float*)d_in[11];
    P.conv1_w    = (const float*)d_in[12];
    P.conv1_b    = (const float*)d_in[13];
    P.x_proj_w   = (const float*)d_in[14];
    P.x_proj1_w  = (const float*)d_in[15];
    P.dt_proj_w  = (const float*)d_in[16];
    P.dt_proj1_w = (const float*)d_in[17];
    P.out_proj_w = (const float*)d_in[18];
    P.A_log      = (const float*)d_in[19];
    P.A_log1     = (const float*)d_in[20];
    P.Dv         = (const float*)d_in[21];
    P.D1v        = (const float*)d_in[22];

    float* ws = (float*)d_ws;
    size_t off = 0;
    auto take = [&](size_t n) { float* p = ws + off; off += n; return p; };

    float* X = take((size_t)M_TOK * D_MODEL);
    float* Y = take((size_t)M_TOK * D_MODEL);
    Bufs W;
    W.xz   = take((size_t)BB * 2 * D_INNER * LL);
    W.x1p  = take((size_t)BB * D_INNER * LL);
    W.co   = take((size_t)BB * D_INNER * LL);
    W.co1  = take((size_t)BB * D_INNER * LL);
    W.xd   = take((size_t)BB * XD_ROWS * LL);
    W.xd1  = take((size_t)BB * XD_ROWS * LL);
    W.dl0  = take((size_t)BB * D_INNER * LL);
    W.dl1  = take((size_t)BB * D_INNER * LL);
    W.y0   = take((size_t)BB * D_INNER * LL);
    W.y1   = take((size_t)BB * D_INNER * LL);
    W.hag0 = take((size_t)BB * D_INNER * NCHUNK * D_STATE);
    W.hag1 = take((size_t)BB * D_INNER * NCHUNK * D_STATE);
    W.sdt0 = take((size_t)BB * D_INNER * NCHUNK);
    W.sdt1 = take((size_t)BB * D_INNER * NCHUNK);

    // x = LN(x_a), y = LN(x_b)
    ln_kernel<false><<<M_TOK / 4, 128, 0, stream>>>(x_a, nullptr, ln1_w, ln1_b, X);
    ln_kernel<false><<<M_TOK / 4, 128, 0, stream>>>(x_b, nullptr, ln2_w, ln2_b, Y);

    // x = relu(cross(x, y, i) + x), i = 0..1
    cross_ssm(X, Y, 0, P, W, stream);
    cross_ssm(X, Y, 1, P, W, stream);
    // y = relu(cross(y, x, 2+i) + y), i = 0..1
    cross_ssm(Y, X, 2, P, W, stream);
    cross_ssm(Y, X, 3, P, W, stream);

    // out = LN(x + y)
    ln_kernel<true><<<M_TOK / 4, 128, 0, stream>>>(X, Y, ln3_w, ln3_b, (float*)d_out);
}